// Head_86019605004616
// MI455X (gfx1250) — compile-verified
//
#include <hip/hip_runtime.h>
#include <hip/hip_bf16.h>
#include <math.h>

typedef __attribute__((ext_vector_type(2))) float v2f;
typedef __attribute__((ext_vector_type(8))) float v8f;
typedef unsigned int u32x4 __attribute__((ext_vector_type(4)));
typedef int          i32x4 __attribute__((ext_vector_type(4)));
typedef int          i32x8 __attribute__((ext_vector_type(8)));

#define B   128
#define DZ  1024
#define L   4096
#define D   256
#define NS  7
#define NT  4            // tiles (16 rows each) per wave in k_content
#define EPSF 1e-8f

// workspace layout (float offsets)
#define OFF_K      0                        // B*D
#define OFF_KNORM  (OFF_K + B*D)            // B
#define OFF_SHIFT  (OFF_KNORM + B)          // B*NS
#define OFF_BETA   (OFF_SHIFT + B*NS)       // B
#define OFF_GATE   (OFF_BETA + B)           // B
#define OFF_GAMMA  (OFF_GATE + B)           // B
#define OFF_SIM    (OFF_GAMMA + B)          // B*L
#define OFF_WI     (OFF_SIM + (size_t)B*L)  // B*L

#if __has_builtin(__builtin_amdgcn_tensor_load_to_lds)
#define HAVE_TDM 1
#warning "CDNA5 probe: TDM builtin AVAILABLE (HAVE_TDM=1)"
#if __has_include(<hip/amd_detail/amd_gfx1250_TDM.h>)
#define TDM_SIX_ARG 1
#warning "CDNA5 probe: using 6-arg tensor_load_to_lds"
#else
#warning "CDNA5 probe: using 5-arg tensor_load_to_lds"
#endif
#else
#warning "CDNA5 probe: TDM builtin MISSING (fallback global loads)"
#endif

// ---------------------------------------------------------------------------
// Kernel 3 (placed first so disasm snippet shows it): 512MB streamer.
// One wave per block; NT=4 tiles of 16x256 f32. TDM DMAs each tile into a
// double-buffered LDS slot (TENSORcnt, 2-deep pipeline) while WMMA computes
// dot(M_row, k) and VALU accumulates ||row||^2 on the previous tile.
// ---------------------------------------------------------------------------
__global__ void __launch_bounds__(32)
k_content(const float* __restrict__ M, float* __restrict__ ws) {
  const int b     = blockIdx.y;
  const int tb    = blockIdx.x;        // block covers NT*16 = 64 rows
  const int lane  = threadIdx.x;       // 0..31
  const int l15   = lane & 15;
  const int half  = lane >> 4;

  __shared__ float kvec[D];
  __shared__ float buf[2][16 * D];     // 2 x 16KB double buffer
  __shared__ float sdot[16];
  __shared__ float sss [16];

  const int rowBase = tb * (NT * 16);
  const float* gbase = M + ((size_t)b * L + rowBase) * D;

#if defined(HAVE_TDM)
  auto tdm_issue = [&](int tileIdx, int slot) {
    const unsigned long long gaddr =
        (unsigned long long)(uintptr_t)(gbase + (size_t)tileIdx * 16 * D);
    const unsigned ldsOff = (unsigned)(uintptr_t)(&buf[slot][0]);
    u32x4 g0;
    g0.x = 1u;                                        // count=1, user D#
    g0.y = ldsOff;                                    // lds_addr (bytes)
    g0.z = (unsigned)(gaddr & 0xFFFFFFFFu);           // global_addr[31:0]
    g0.w = (unsigned)((gaddr >> 32) & 0x01FFFFFFu)    // global_addr[56:32]
         | (2u << 30);                                // type=2 (image)
    i32x8 g1;
    g1[0] = 0x00020000;                // wg_mask=0, data_size=2 (4 bytes)
    g1[1] = (int)(256u << 16);         // tensor_dim0=256 (bits 79:48 low part)
    g1[2] = 0;
    g1[3] = (int)((256u << 16) | 16u); // tensor_dim1 hi16=16, tile_dim0=256
    g1[4] = 16;                        // tile_dim1=16 rows
    g1[5] = 256;                       // tensor_dim0_stride=256 elems
    g1[6] = 0;
    g1[7] = 0;
    i32x4 gz; gz[0] = 0; gz[1] = 0; gz[2] = 0; gz[3] = 0;
#if defined(TDM_SIX_ARG)
    i32x8 gz8; for (int i = 0; i < 8; ++i) gz8[i] = 0;
    __builtin_amdgcn_tensor_load_to_lds(g0, g1, gz, gz, gz8, 0);
#else
    __builtin_amdgcn_tensor_load_to_lds(g0, g1, gz, gz, 0);
#endif
  };
  tdm_issue(0, 0);
  tdm_issue(1, 1);
#endif

  // stage k while DMA is in flight
  for (int i = lane; i < D; i += 32) kvec[i] = ws[OFF_K + (size_t)b * D + i];
  __syncthreads();
  const float knorm = ws[OFF_KNORM + b];
  const float beta  = ws[OFF_BETA + b];

#pragma unroll
  for (int it = 0; it < NT; ++it) {
#if defined(HAVE_TDM)
    if (it < NT - 1) __builtin_amdgcn_s_wait_tensorcnt(1);  // tile it resident
    else             __builtin_amdgcn_s_wait_tensorcnt(0);
    const float* arow = &buf[it & 1][l15 * D];
#else
    const float* arow = gbase + (size_t)(it * 16 + l15) * D;
    __builtin_prefetch(arow, 0, 0);
#endif

    v8f c = {};
    float ssq = 0.f;
    for (int kc = 0; kc < D; kc += 4) {
      const int k0 = kc + 2 * half;
      v2f a, bb;
      a.x  = arow[k0];
      a.y  = arow[k0 + 1];
      bb.x = kvec[k0];
      bb.y = kvec[k0 + 1];
      ssq = fmaf(a.x, a.x, ssq);
      ssq = fmaf(a.y, a.y, ssq);
      c = __builtin_amdgcn_wmma_f32_16x16x4_f32(false, a, false, bb,
                                                (short)0, c, false, false);
    }

#if defined(HAVE_TDM)
    if (it + 2 < NT) tdm_issue(it + 2, it & 1);   // refill freed slot
#endif

    ssq += __shfl_xor(ssq, 16);        // combine lane m with lane m+16 (wave32)

    // column 0 of D holds dot(row_m, k): lane 0 -> rows 0..7, lane 16 -> 8..15
    if (l15 == 0) {
#pragma unroll
      for (int i = 0; i < 8; ++i) sdot[half * 8 + i] = c[i];
    }
    if (half == 0) sss[l15] = ssq;
    __syncthreads();

    if (half == 0) {
      const int m = l15;
      const float dot = sdot[m];
      const float mn  = sqrtf(sss[m]);
      const float sim = dot / (knorm * mn + EPSF);
      ws[OFF_SIM + (size_t)b * L + rowBase + it * 16 + m] = beta * sim;
    }
    __syncthreads();
  }
}

// ---------------------------------------------------------------------------
// Kernel 1: k = tanh(Z @ key_W + key_b) via f32 WMMA (16x16 tile per wave)
// ---------------------------------------------------------------------------
__global__ void __launch_bounds__(32)
k_heads_gemm(const float* __restrict__ Z, const float* __restrict__ keyW,
             const float* __restrict__ keyB, float* __restrict__ kout) {
  const int lane = threadIdx.x;
  const int l15  = lane & 15;
  const int half = lane >> 4;
  const int colBase = blockIdx.x * 16;   // over D
  const int rowBase = blockIdx.y * 16;   // over B
  const float* zrow = Z + (size_t)(rowBase + l15) * DZ;
  v8f c = {};
  for (int kk = 0; kk < DZ; kk += 4) {
    const int k0 = kk + 2 * half;
    v2f a, bb;
    a.x  = zrow[k0];
    a.y  = zrow[k0 + 1];
    bb.x = keyW[(size_t)k0 * D + colBase + l15];
    bb.y = keyW[(size_t)(k0 + 1) * D + colBase + l15];
    c = __builtin_amdgcn_wmma_f32_16x16x4_f32(false, a, false, bb,
                                              (short)0, c, false, false);
  }
#pragma unroll
  for (int i = 0; i < 8; ++i) {
    const int m = i + 8 * half;
    const int n = l15;
    kout[(size_t)(rowBase + m) * D + colBase + n] =
        tanhf(c[i] + keyB[colBase + n]);
  }
}

// ---------------------------------------------------------------------------
// Kernel 2: shift softmax, beta/gate/gamma heads, and ||k|| per batch row
// ---------------------------------------------------------------------------
__global__ void __launch_bounds__(256)
k_heads_small(const float* __restrict__ Z,
              const float* __restrict__ shiftW, const float* __restrict__ shiftB,
              const float* __restrict__ betaW,  const float* __restrict__ betaB,
              const float* __restrict__ gateW,  const float* __restrict__ gateB,
              const float* __restrict__ gammaW, const float* __restrict__ gammaB,
              float* __restrict__ ws) {
  const int b = blockIdx.x;
  const int t = threadIdx.x;
  __shared__ float red[256];
  __shared__ float res[NS + 3];
  const float* zrow = Z + (size_t)b * DZ;

  float acc[NS + 3];
#pragma unroll
  for (int j = 0; j < NS + 3; ++j) acc[j] = 0.f;
  for (int i = t; i < DZ; i += 256) {
    const float z = zrow[i];
#pragma unroll
    for (int n = 0; n < NS; ++n) acc[n] = fmaf(z, shiftW[(size_t)i * NS + n], acc[n]);
    acc[NS + 0] = fmaf(z, betaW[i],  acc[NS + 0]);
    acc[NS + 1] = fmaf(z, gateW[i],  acc[NS + 1]);
    acc[NS + 2] = fmaf(z, gammaW[i], acc[NS + 2]);
  }
  for (int j = 0; j < NS + 3; ++j) {
    red[t] = acc[j];
    __syncthreads();
    for (int s = 128; s > 0; s >>= 1) { if (t < s) red[t] += red[t + s]; __syncthreads(); }
    if (t == 0) res[j] = red[0];
    __syncthreads();
  }
  {
    const float kv = ws[OFF_K + (size_t)b * D + t];
    red[t] = kv * kv;
  }
  __syncthreads();
  for (int s = 128; s > 0; s >>= 1) { if (t < s) red[t] += red[t + s]; __syncthreads(); }

  if (t == 0) {
    float sv[NS], mx = -1e30f, ssum = 0.f;
#pragma unroll
    for (int n = 0; n < NS; ++n) { sv[n] = res[n] + shiftB[n]; mx = fmaxf(mx, sv[n]); }
#pragma unroll
    for (int n = 0; n < NS; ++n) { sv[n] = expf(sv[n] - mx); ssum += sv[n]; }
#pragma unroll
    for (int n = 0; n < NS; ++n) ws[OFF_SHIFT + b * NS + n] = sv[n] / ssum;
    ws[OFF_BETA  + b] = log1pf(expf(res[NS + 0] + betaB[0]));
    ws[OFF_GATE  + b] = 1.f / (1.f + expf(-(res[NS + 1] + gateB[0])));
    ws[OFF_GAMMA + b] = log1pf(expf(res[NS + 2] + gammaB[0])) + 1.f;
    ws[OFF_KNORM + b] = sqrtf(red[0]);
  }
}

// ---------------------------------------------------------------------------
// Kernel 4: per-batch softmax over L, then gate interpolation with W_old
// ---------------------------------------------------------------------------
__global__ void __launch_bounds__(1024)
k_softmax_interp(const float* __restrict__ W_old, float* __restrict__ ws) {
  const int b = blockIdx.x;
  const int t = threadIdx.x;
  __shared__ float red[1024];
  const float* srow = ws + OFF_SIM + (size_t)b * L;

  float v[4];
  float mx = -1e30f;
#pragma unroll
  for (int i = 0; i < 4; ++i) { v[i] = srow[t + i * 1024]; mx = fmaxf(mx, v[i]); }
  red[t] = mx; __syncthreads();
  for (int s = 512; s > 0; s >>= 1) { if (t < s) red[t] = fmaxf(red[t], red[t + s]); __syncthreads(); }
  mx = red[0];
  __syncthreads();

  float sum = 0.f;
#pragma unroll
  for (int i = 0; i < 4; ++i) { v[i] = expf(v[i] - mx); sum += v[i]; }
  red[t] = sum; __syncthreads();
  for (int s = 512; s > 0; s >>= 1) { if (t < s) red[t] += red[t + s]; __syncthreads(); }
  sum = red[0];

  const float g = ws[OFF_GATE + b];
  float* wrow = ws + OFF_WI + (size_t)b * L;
  const float* worow = W_old + (size_t)b * L;
#pragma unroll
  for (int i = 0; i < 4; ++i) {
    const int l = t + i * 1024;
    wrow[l] = fmaf(g, worow[l], (1.f - g) * (v[i] / sum));
  }
}

// ---------------------------------------------------------------------------
// Kernel 5: circular 7-tap shift convolution, sharpen (pow gamma), normalize
// ---------------------------------------------------------------------------
__global__ void __launch_bounds__(1024)
k_shift_sharpen(float* __restrict__ ws, float* __restrict__ out) {
  const int b = blockIdx.x;
  const int t = threadIdx.x;
  __shared__ float red[1024];
  __shared__ float sh[NS];
  if (t < NS) sh[t] = ws[OFF_SHIFT + b * NS + t];
  __syncthreads();

  const float* wrow = ws + OFF_WI + (size_t)b * L;
  const float gamma = ws[OFF_GAMMA + b];

  float v[4];
  float sum = 0.f;
#pragma unroll
  for (int i = 0; i < 4; ++i) {
    const int l = t + i * 1024;
    float acc = 0.f;
#pragma unroll
    for (int n = 0; n < NS; ++n) {
      const int idx = (l - (n - NS / 2)) & (L - 1);
      acc = fmaf(sh[n], wrow[idx], acc);
    }
    const float p = powf(fmaxf(acc, 0.f), gamma);
    v[i] = p;
    sum += p;
  }
  red[t] = sum; __syncthreads();
  for (int s = 512; s > 0; s >>= 1) { if (t < s) red[t] += red[t + s]; __syncthreads(); }
  sum = red[0];

#pragma unroll
  for (int i = 0; i < 4; ++i) {
    const int l = t + i * 1024;
    out[(size_t)b * L + l] = v[i] / sum;
  }
}

// ---------------------------------------------------------------------------
extern "C" void kernel_launch(void* const* d_in, const int* in_sizes, int n_in,
                              void* d_out, int out_size, void* d_ws, size_t ws_size,
                              hipStream_t stream) {
  const float* Z      = (const float*)d_in[0];
  const float* W_old  = (const float*)d_in[1];
  const float* M      = (const float*)d_in[2];
  const float* keyW   = (const float*)d_in[3];
  const float* keyB   = (const float*)d_in[4];
  const float* shiftW = (const float*)d_in[5];
  const float* shiftB = (const float*)d_in[6];
  const float* betaW  = (const float*)d_in[7];
  const float* betaB  = (const float*)d_in[8];
  const float* gateW  = (const float*)d_in[9];
  const float* gateB  = (const float*)d_in[10];
  const float* gammaW = (const float*)d_in[11];
  const float* gammaB = (const float*)d_in[12];
  float* ws  = (float*)d_ws;
  float* out = (float*)d_out;

  k_heads_gemm<<<dim3(D / 16, B / 16), 32, 0, stream>>>(Z, keyW, keyB, ws + OFF_K);
  k_heads_small<<<dim3(B), 256, 0, stream>>>(Z, shiftW, shiftB, betaW, betaB,
                                             gateW, gateB, gammaW, gammaB, ws);
  k_content<<<dim3(L / (NT * 16), B), 32, 0, stream>>>(M, ws);
  k_softmax_interp<<<dim3(B), 1024, 0, stream>>>(W_old, ws);
  k_shift_sharpen<<<dim3(B), 1024, 0, stream>>>(ws, out);
}